// ISGPModel_25726854103108
// MI455X (gfx1250) — compile-verified
//
#include <hip/hip_runtime.h>
#include <math.h>

typedef __attribute__((ext_vector_type(2))) float v2f;
typedef __attribute__((ext_vector_type(8))) float v8f;

#define B_    16
#define D_    67
#define H_    128
#define W_    128
#define HWc   (H_ * W_)
#define FD_   64
#define L_    128
#define N_    32
#define HID_  64
#define EPS2_ 0.01f
#define SCALE_ 0.125f   // sqrt(2/128)

// Workspace layout per batch (floats)
#define WS_SZH    0
#define WS_ZR     2048
#define WS_V      2144
#define WS_NRM    2176
#define WS_NR2    2208
#define WS_STRIDE 2240

__device__ __forceinline__ float softplusf(float x) {
  return (x > 20.f) ? x : log1pf(__expf(x));
}

__device__ __forceinline__ v8f wmma4(v2f a, v2f b, v8f c) {
  // V_WMMA_F32_16X16X4_F32: D = A(16x4) * B(4x16) + C(16x16)
  return __builtin_amdgcn_wmma_f32_16x16x4_f32(
      /*neg_a=*/false, a, /*neg_b=*/false, b,
      /*c_mod=*/(short)0, c, /*reuse_a=*/false, /*reuse_b=*/false);
}

// ---------------------------------------------------------------------------
// Kernel A: per-batch setup. Gathers the 32 inducing points, builds
// Kmm + jitter, solves (Kmm+eps I) v = m - phi@omega with Gauss-Jordan
// (matrix is SPD: RBF kernels + 0.01 I), writes szh/zr/v/nrm/nr2 to ws.
// ---------------------------------------------------------------------------
__global__ __launch_bounds__(256) void isgp_setup(
    const float* __restrict__ feature, const float* __restrict__ theta_w,
    const float* __restrict__ theta_b, const float* __restrict__ omega,
    const float* __restrict__ logsigma2, const float* __restrict__ weights,
    const float* __restrict__ u_w1, const float* __restrict__ u_b1,
    const float* __restrict__ u_w2, const float* __restrict__ u_b2,
    const int* __restrict__ points, float* __restrict__ ws) {
  __shared__ float zf[N_][68];
  __shared__ float sig[FD_];
  __shared__ float aug[N_][33];
  __shared__ float po[N_], macc[N_], nrmS[N_], nr2S[N_], zrS[N_ * 3];
  __shared__ float wgtS;

  const int b = blockIdx.x;
  const int t = threadIdx.x;
  float* wb = ws + (size_t)b * WS_STRIDE;

  if (t < FD_) sig[t] = __expf(logsigma2[t]);
  if (t == 0) wgtS = softplusf(weights[0]);
  for (int idx = t; idx < N_ * D_; idx += 256) {
    int n = idx / D_, d = idx % D_;
    int y = points[(b * N_ + n) * 2 + 0];
    int x = points[(b * N_ + n) * 2 + 1];
    zf[n][d] = feature[(((size_t)b * D_ + d) * H_ + y) * W_ + x];
  }
  __syncthreads();

  if (t < N_) {
    float s = 0.f;
    for (int d = 0; d < FD_; ++d) s += sig[d] * zf[t][d] * zf[t][d];
    nrmS[t] = 0.5f * s;
    float r = zf[t][64] * zf[t][64] + zf[t][65] * zf[t][65] + zf[t][66] * zf[t][66];
    nr2S[t] = 0.5f * r;
    zrS[t * 3 + 0] = zf[t][64];
    zrS[t * 3 + 1] = zf[t][65];
    zrS[t * 3 + 2] = zf[t][66];
    po[t] = 0.f;
    macc[t] = 0.f;
    wb[WS_NRM + t] = nrmS[t];
    wb[WS_NR2 + t] = nr2S[t];
  }
  __syncthreads();

  // szh = sig * zh  -> workspace (pixel kernel's GEMM2 A-matrix)
  for (int idx = t; idx < N_ * FD_; idx += 256) {
    int n = idx / FD_, d = idx % FD_;
    wb[WS_SZH + idx] = sig[d] * zf[n][d];
  }
  if (t < N_ * 3) wb[WS_ZR + t] = zrS[t];

  // Kmm + jitter -> augmented matrix
  for (int idx = t; idx < N_ * N_; idx += 256) {
    int i = idx / N_, j = idx % N_;
    float s = 0.f;
    for (int d = 0; d < FD_; ++d) s += sig[d] * zf[i][d] * zf[j][d];
    float r2 = 0.f;
    for (int k = 0; k < 3; ++k) {
      float df = zrS[i * 3 + k] - zrS[j * 3 + k];
      r2 += df * df;
    }
    float K = __expf(s - nrmS[i] - nrmS[j]) + wgtS * __expf(-0.5f * r2);
    aug[i][j] = K + ((i == j) ? EPS2_ : 0.f);
  }

  // po[n] = sum_l scale*sin(zf[n].theta_w[l] + theta_b[l]) * omega[l]
  {
    int n = t >> 3;
    int l0 = (t & 7) * 16;
    float s = 0.f;
    for (int l = l0; l < l0 + 16; ++l) {
      float lin = theta_b[l];
      for (int d = 0; d < D_; ++d) lin += zf[n][d] * theta_w[l * D_ + d];
      s += SCALE_ * __sinf(lin) * omega[l];
    }
    atomicAdd(&po[n], s);
  }

  // macc[n] = sum_h relu(zf[n].u_w1[h] + u_b1[h]) * u_w2[h]
  {
    int n = t >> 3;
    int h0 = (t & 7) * 8;
    float s = 0.f;
    for (int h = h0; h < h0 + 8; ++h) {
      float a = u_b1[h];
      for (int d = 0; d < D_; ++d) a += zf[n][d] * u_w1[h * D_ + d];
      if (a > 0.f) s += a * u_w2[h];
    }
    atomicAdd(&macc[n], s);
  }
  __syncthreads();

  if (t < N_) {
    float m = softplusf(macc[t] + u_b2[0]) * ((t < N_ / 2) ? 1.f : -1.f);
    aug[t][32] = m - po[t];  // rhs
  }
  __syncthreads();

  // Gauss-Jordan elimination (SPD, no pivoting needed)
  for (int k = 0; k < N_; ++k) {
    if (t < N_ && t != k) {
      float fk = aug[t][k] / aug[k][k];
      for (int j = k; j <= 32; ++j) aug[t][j] -= fk * aug[k][j];
    }
    __syncthreads();
  }
  if (t < N_) wb[WS_V + t] = aug[t][32] / aug[t][t];
}

// ---------------------------------------------------------------------------
// Kernel B: pixel kernel. Each wave owns 16-pixel groups (contiguous in w).
// Per k-step (4 d-slices): one B-fragment load feeds 8 theta WMMAs (L=128)
// + 2 szh WMMAs (N=32) + the n1 norm accumulation. Epilogue: sin/exp,
// cross-half reduce via shfl_xor(16), coalesced store.
// waves_per_eu(4) caps VGPRs at 256: no s_set_vgpr_msb churn, 4 waves/SIMD.
// ---------------------------------------------------------------------------
__global__ __launch_bounds__(128)
__attribute__((amdgpu_waves_per_eu(4))) void isgp_pixel(
    const float* __restrict__ feature, const float* __restrict__ theta_w,
    const float* __restrict__ theta_b, const float* __restrict__ omega,
    const float* __restrict__ logsigma2, const float* __restrict__ weights,
    const float* __restrict__ ws, float* __restrict__ out) {
  __shared__ float thetaS[L_ * 68];  // theta_w padded to 68 cols (col 67 = 0)
  __shared__ float szhS[N_ * 68];    // sig*zh padded rows (stride 68)
  __shared__ float omegaS[L_], thbS[L_], sigS[FD_];
  __shared__ float vS[N_], nrmS[N_], nr2S[N_], zrS[N_ * 3];
  __shared__ float wgtS;

  const int b = blockIdx.x >> 4;   // batch
  const int ib = blockIdx.x & 15;  // block-within-batch
  const int t = threadIdx.x;
  const float* wb = ws + (size_t)b * WS_STRIDE;

  for (int idx = t; idx < L_ * 68; idx += 128) {
    int l = idx / 68, c = idx % 68;
    thetaS[idx] = (c < D_) ? theta_w[l * D_ + c] : 0.f;
  }
  for (int idx = t; idx < N_ * FD_; idx += 128) {
    int n = idx / FD_, d = idx % FD_;
    szhS[n * 68 + d] = wb[WS_SZH + idx];
  }
  if (t < L_) { omegaS[t] = omega[t]; thbS[t] = theta_b[t]; }
  if (t < FD_) sigS[t] = __expf(logsigma2[t]);
  if (t < N_) { vS[t] = wb[WS_V + t]; nrmS[t] = wb[WS_NRM + t]; nr2S[t] = wb[WS_NR2 + t]; }
  if (t < N_ * 3) zrS[t] = wb[WS_ZR + t];
  if (t == 0) wgtS = softplusf(weights[0]);
  __syncthreads();

  const int wv = t >> 5;
  const int lane = t & 31;
  const int half = lane >> 4;   // C/D tile: rows M = i + 8*half; B tile: K = i + 2*half
  const int pcol = lane & 15;   // pixel column within group / A-matrix row M
  const float* fb = feature + (size_t)b * D_ * HWc;
  float* ob = out + (size_t)b * HWc;

  for (int it = 0; it < 16; ++it) {
    const int gid = (ib * 4 + wv) * 16 + it;  // 0..1023 group within batch
    const int pbase = gid * 16;
    const float* fp = fb + pbase + pcol;

    v8f accT[8] = {};  // theta GEMM: 8 L-tiles of 16
    v8f accZ[2] = {};  // szh GEMM: 2 n-tiles of 16
    float n1p = 0.f;

#pragma unroll 2
    for (int k = 0; k < 16; ++k) {
      const int d0 = k * 4 + 2 * half;
      v2f bf;
      bf.x = __builtin_nontemporal_load(&fp[d0 * HWc]);
      bf.y = __builtin_nontemporal_load(&fp[(d0 + 1) * HWc]);
      n1p += sigS[d0] * bf.x * bf.x + sigS[d0 + 1] * bf.y * bf.y;
#pragma unroll
      for (int lt = 0; lt < 8; ++lt) {
        v2f af = *(const v2f*)&thetaS[(lt * 16 + pcol) * 68 + d0];
        accT[lt] = wmma4(af, bf, accT[lt]);
      }
#pragma unroll
      for (int nt = 0; nt < 2; ++nt) {
        v2f af = *(const v2f*)&szhS[(nt * 16 + pcol) * 68 + d0];
        accZ[nt] = wmma4(af, bf, accZ[nt]);
      }
    }
    float n1 = 0.5f * (n1p + __shfl_xor(n1p, 16, 32));

    // Remainder d = 64..66 for the theta GEMM (K row 3 zero-padded),
    // plus nr1 = 0.5 * (f64^2 + f65^2 + f66^2)
    float nr1;
    const float fr0 = __builtin_nontemporal_load(&fp[64 * HWc]);
    const float fr1 = __builtin_nontemporal_load(&fp[65 * HWc]);
    const float fr2 = __builtin_nontemporal_load(&fp[66 * HWc]);
    {
      const int d0 = 64 + 2 * half;  // half0: d64,d65 ; half1: d66, pad
      v2f bf;
      bf.x = (half == 0) ? fr0 : fr2;
      bf.y = (half == 0) ? fr1 : 0.f;
      float nr1p = 0.5f * (bf.x * bf.x + bf.y * bf.y);
      nr1 = nr1p + __shfl_xor(nr1p, 16, 32);
#pragma unroll
      for (int lt = 0; lt < 8; ++lt) {
        v2f af = *(const v2f*)&thetaS[(lt * 16 + pcol) * 68 + d0];
        accT[lt] = wmma4(af, bf, accT[lt]);
      }
    }

    // prior = scale * sum_l sin(lin_l + theta_b_l) * omega_l
    float pr = 0.f;
#pragma unroll
    for (int lt = 0; lt < 8; ++lt) {
#pragma unroll
      for (int i = 0; i < 8; ++i) {
        int l = lt * 16 + 8 * half + i;
        pr += __sinf(accT[lt][i] + thbS[l]) * omegaS[l];
      }
    }
    pr += __shfl_xor(pr, 16, 32);

    // update = sum_n v_n * ( exp(a_n - nrm_n - n1) + wgt*exp(fr.zr_n - nr2_n - nr1) )
    float up = 0.f;
#pragma unroll
    for (int nt = 0; nt < 2; ++nt) {
#pragma unroll
      for (int i = 0; i < 8; ++i) {
        int n = nt * 16 + 8 * half + i;
        float brd = fr0 * zrS[n * 3] + fr1 * zrS[n * 3 + 1] + fr2 * zrS[n * 3 + 2];
        float e = __expf(accZ[nt][i] - nrmS[n] - n1) +
                  wgtS * __expf(brd - nr2S[n] - nr1);
        up += vS[n] * e;
      }
    }
    up += __shfl_xor(up, 16, 32);

    if (half == 0) {
      __builtin_nontemporal_store(SCALE_ * pr + up, &ob[pbase + pcol]);
    }
  }
}

extern "C" void kernel_launch(void* const* d_in, const int* in_sizes, int n_in,
                              void* d_out, int out_size, void* d_ws, size_t ws_size,
                              hipStream_t stream) {
  const float* feature   = (const float*)d_in[0];
  const float* theta_w   = (const float*)d_in[1];
  const float* theta_b   = (const float*)d_in[2];
  const float* omega     = (const float*)d_in[3];
  const float* logsigma2 = (const float*)d_in[4];
  const float* weights   = (const float*)d_in[5];
  const float* u_w1      = (const float*)d_in[6];
  const float* u_b1      = (const float*)d_in[7];
  const float* u_w2      = (const float*)d_in[8];
  const float* u_b2      = (const float*)d_in[9];
  const int*   points    = (const int*)d_in[10];
  float* ws  = (float*)d_ws;
  float* out = (float*)d_out;

  isgp_setup<<<B_, 256, 0, stream>>>(feature, theta_w, theta_b, omega, logsigma2,
                                     weights, u_w1, u_b1, u_w2, u_b2, points, ws);
  isgp_pixel<<<B_ * 16, 128, 0, stream>>>(feature, theta_w, theta_b, omega,
                                          logsigma2, weights, ws, out);
}